// TopologicallyDenseRegularization_44727789421016
// MI455X (gfx1250) — compile-verified
//
#include <hip/hip_runtime.h>
#include <hip/hip_bf16.h>
#include <cstdint>

typedef float v2f __attribute__((ext_vector_type(2)));
typedef float v8f __attribute__((ext_vector_type(8)));

#define NCLS 8
#define NPTS 2048
#define DIM 64
#define TPR (NPTS / 16)          // 128 tiles per row
#define TPC (TPR * TPR)          // 16384 tiles per class

// ---------------------------------------------------------------------------
// Stage 0: squared row norms (and zero the scalar output for this call).
// ---------------------------------------------------------------------------
__global__ void sqnorm_kernel(const float* __restrict__ X,
                              float* __restrict__ sq,
                              float* __restrict__ out) {
    int i = blockIdx.x * blockDim.x + threadIdx.x;   // 0 .. NCLS*NPTS-1
    if (i == 0) out[0] = 0.0f;
    const float* xi = X + (size_t)i * DIM;
    float s = 0.0f;
#pragma unroll 8
    for (int k = 0; k < DIM; ++k) s = fmaf(xi[k], xi[k], s);
    sq[i] = s;
}

// ---------------------------------------------------------------------------
// Stage 1: D[c] = pairwise distances via WMMA f32 16x16x4 Gram tiles.
// One wave per 16x16 output tile; 8 waves per 256-thread block.
// ---------------------------------------------------------------------------
__global__ void gram_dist_kernel(const float* __restrict__ X,
                                 const float* __restrict__ sq,
                                 float* __restrict__ D) {
    const int wave_g = (blockIdx.x * blockDim.x + threadIdx.x) >> 5;
    const int lane   = threadIdx.x & 31;

    const int c  = wave_g / TPC;
    const int t  = wave_g % TPC;
    const int mt = t / TPR;
    const int nt = t % TPR;

    const float* Xc = X + (size_t)c * NPTS * DIM;
    const int r  = lane & 15;           // row within tile (A: M, B: N)
    const int kp = (lane >> 4) << 1;    // K sub-pair selected by lane half

    const float* arow = Xc + (size_t)(mt * 16 + r) * DIM + kp;
    const float* brow = Xc + (size_t)(nt * 16 + r) * DIM + kp;  // B = X^T tile

    v8f acc = {};
#pragma unroll
    for (int k = 0; k < DIM; k += 4) {
        v2f a = { arow[k], arow[k + 1] };
        v2f b = { brow[k], brow[k + 1] };
        // D = A x B + C ; 16x16x4 f32, EXEC is all-ones here (full waves only)
        acc = __builtin_amdgcn_wmma_f32_16x16x4_f32(
            /*neg_a=*/false, a, /*neg_b=*/false, b,
            /*c_mod=*/(short)0, acc, /*reuse_a=*/false, /*reuse_b=*/false);
    }

    // C/D layout: VGPR rr, lanes 0-15 -> M = rr, lanes 16-31 -> M = rr + 8
    const float* sqc  = sq + c * NPTS;
    const int n_col   = nt * 16 + (lane & 15);
    const float sqn   = sqc[n_col];
    const int   mbase = mt * 16 + ((lane >> 4) << 3);
    float* Dc = D + (size_t)c * NPTS * NPTS;
#pragma unroll
    for (int rr = 0; rr < 8; ++rr) {
        const int m = mbase + rr;
        float d2 = fmaxf(sqc[m] + sqn - 2.0f * acc[rr], 0.0f);
        float d  = (d2 > 0.0f) ? sqrtf(d2) : 0.0f;
        Dc[(size_t)m * NPTS + n_col] = d;
    }
}

// ---------------------------------------------------------------------------
// Stage 2: Prim's MST per class (one block per class), rows streamed from L2.
// key[i] == +INF  <=>  vertex i is in the tree (all true distances finite).
// ---------------------------------------------------------------------------
__global__ void prim_kernel(const float* __restrict__ D,
                            float* __restrict__ out) {
    __shared__ float key[NPTS];
    __shared__ unsigned long long minpack;

    const int c   = blockIdx.x;
    const int tid = threadIdx.x;            // 256 threads, 8 waves
    const float* Dc = D + (size_t)c * NPTS * NPTS;
    const float INF = __builtin_inff();

    // init: key = dist row 0, vertex 0 starts in the tree
    for (int i = tid; i < NPTS; i += blockDim.x)
        key[i] = (i == 0) ? INF : Dc[i];
    __syncthreads();

    float sum = 0.0f;
    for (int it = 0; it < NPTS - 1; ++it) {
        if (tid == 0) minpack = ~0ULL;
        __syncthreads();

        // argmin over keys, packed so ties resolve to the lowest index
        unsigned long long best = ~0ULL;
#pragma unroll
        for (int s = 0; s < NPTS / 256; ++s) {
            const int i = tid + s * 256;
            unsigned long long p =
                ((unsigned long long)__float_as_uint(key[i]) << 32) | (unsigned)i;
            best = (p < best) ? p : best;
        }
        atomicMin(&minpack, best);          // ds_min_u64 on LDS
        __syncthreads();

        const unsigned long long mp = minpack;
        const int   j   = (int)(mp & 0xffffffffu);
        const float val = __uint_as_float((unsigned)(mp >> 32));
        sum += fabsf(val - 1.0f);           // tracked redundantly; tid 0 writes

        // relax against row j (8 KB, L2-resident); owner thread retires j
        const float* rowj = Dc + (size_t)j * NPTS;
#pragma unroll
        for (int s = 0; s < NPTS / 256; ++s) {
            const int i = tid + s * 256;
            const float kv = key[i];
            key[i] = (kv == INF || i == j) ? INF : fminf(kv, rowj[i]);
        }
        __syncthreads();
    }

    if (tid == 0) atomicAdd(out, sum / (float)(NPTS - 1));
}

// ---------------------------------------------------------------------------
extern "C" void kernel_launch(void* const* d_in, const int* in_sizes, int n_in,
                              void* d_out, int out_size, void* d_ws, size_t ws_size,
                              hipStream_t stream) {
    const float* X = (const float*)d_in[0];           // [8*2048, 64] f32
    float* out = (float*)d_out;                       // scalar f32

    float* sq = (float*)d_ws;                                      // 64 KB
    float* D  = (float*)((char*)d_ws + (size_t)NCLS * NPTS * 4);   // 128 MB

    // Stage 0: norms + zero output
    sqnorm_kernel<<<(NCLS * NPTS) / 256, 256, 0, stream>>>(X, sq, out);

    // Stage 1: WMMA Gram -> distances. 8 waves/block, one tile per wave.
    const int total_waves = NCLS * TPC;               // 131072
    gram_dist_kernel<<<total_waves / 8, 256, 0, stream>>>(X, sq, D);

    // Stage 2: 8 independent Prim runs
    prim_kernel<<<NCLS, 256, 0, stream>>>(D, out);
}